// WTALIF_44143673868827
// MI455X (gfx1250) — compile-verified
//
#include <hip/hip_runtime.h>
#include <cstdint>

// ---------------- problem constants (match reference) ----------------
#define TS      5          // TIME_STEP
#define BS      32         // batch per step = 160/5
#define NB      160        // total batch
#define CP      294912     // features per sample = 128*48*48
#define N4      (CP / 4)   // float4 per sample = 73728
#define KTOP    58982      // int(0.2 * 294912)
#define BINS    65536
#define HWORDS  (BINS / 2) // u16-packed histogram words (128 KB LDS)
#define NCHUNK  1024       // 64 bins per chunk

typedef __attribute__((ext_vector_type(4))) float v4f;

// Order-preserving float->u32 key: larger key <=> larger float.
__device__ __forceinline__ uint32_t fkey(float f) {
    uint32_t u = __float_as_uint(f);
    return (u & 0x80000000u) ? ~u : (u | 0x80000000u);
}

__device__ __forceinline__ void hist_bump(unsigned* hist, uint32_t bin) {
    // bin in [0,65536): packed u16 counters, two bins per 32-bit word
    atomicAdd(&hist[bin >> 1], (bin & 1u) ? 0x10000u : 1u);
}

// ---------------------------------------------------------------------
// Pass 1: per-sample histogram of the top 16 key bits; find the bucket
// holding the K-th largest key and the count strictly greater than it.
// ---------------------------------------------------------------------
__global__ __launch_bounds__(1024) void wta_hist_hi(const float* __restrict__ x,
                                                    int* __restrict__ qArr,
                                                    unsigned* __restrict__ cgArr) {
    __shared__ unsigned hist[HWORDS];
    __shared__ unsigned csum[NCHUNK];
    const int tid = threadIdx.x;

    for (int i = tid; i < HWORDS; i += 1024) hist[i] = 0u;
    __syncthreads();

    const v4f* xv = (const v4f*)(x + (size_t)blockIdx.x * CP);
    for (int i = tid; i < N4; i += 1024) {
        __builtin_prefetch(xv + i + 1024, 0, 0);   // global_prefetch_b8
        v4f v = xv[i];
        hist_bump(hist, fkey(v.x) >> 16);
        hist_bump(hist, fkey(v.y) >> 16);
        hist_bump(hist, fkey(v.z) >> 16);
        hist_bump(hist, fkey(v.w) >> 16);
    }
    __syncthreads();

    // chunk sums: chunk tid covers bins [tid*64, tid*64+64) = words [tid*32, +32)
    {
        unsigned s = 0;
        const int w0 = tid * 32;
        #pragma unroll 8
        for (int w = 0; w < 32; ++w) {
            unsigned h = hist[w0 + w];
            s += (h & 0xFFFFu) + (h >> 16);
        }
        csum[tid] = s;
    }
    __syncthreads();

    if (tid == 0) {
        unsigned cum = 0, cg = 0;
        int q = 0;
        for (int c = NCHUNK - 1; c >= 0; --c) {
            unsigned cs = csum[c];
            if (cum + cs >= KTOP) {
                for (int b = c * 64 + 63; b >= c * 64; --b) {
                    unsigned h = hist[b >> 1];
                    unsigned cnt = (b & 1) ? (h >> 16) : (h & 0xFFFFu);
                    if (cum + cnt >= KTOP) { q = b; cg = cum; break; }
                    cum += cnt;
                }
                break;
            }
            cum += cs;
        }
        qArr[blockIdx.x] = q;
        cgArr[blockIdx.x] = cg;
    }
}

// ---------------------------------------------------------------------
// Pass 2: histogram low 16 key bits of elements inside the candidate
// bucket; resolve the exact 32-bit threshold key per sample.
// ---------------------------------------------------------------------
__global__ __launch_bounds__(1024) void wta_hist_lo(const float* __restrict__ x,
                                                    const int* __restrict__ qArr,
                                                    const unsigned* __restrict__ cgArr,
                                                    unsigned* __restrict__ thrArr) {
    __shared__ unsigned hist[HWORDS];
    __shared__ unsigned csum[NCHUNK];
    const int tid = threadIdx.x;

    for (int i = tid; i < HWORDS; i += 1024) hist[i] = 0u;
    __syncthreads();

    const unsigned q  = (unsigned)qArr[blockIdx.x];
    const unsigned cg = cgArr[blockIdx.x];

    const v4f* xv = (const v4f*)(x + (size_t)blockIdx.x * CP);
    for (int i = tid; i < N4; i += 1024) {
        v4f v = xv[i];
        uint32_t k;
        k = fkey(v.x); if ((k >> 16) == q) hist_bump(hist, k & 0xFFFFu);
        k = fkey(v.y); if ((k >> 16) == q) hist_bump(hist, k & 0xFFFFu);
        k = fkey(v.z); if ((k >> 16) == q) hist_bump(hist, k & 0xFFFFu);
        k = fkey(v.w); if ((k >> 16) == q) hist_bump(hist, k & 0xFFFFu);
    }
    __syncthreads();

    {
        unsigned s = 0;
        const int w0 = tid * 32;
        #pragma unroll 8
        for (int w = 0; w < 32; ++w) {
            unsigned h = hist[w0 + w];
            s += (h & 0xFFFFu) + (h >> 16);
        }
        csum[tid] = s;
    }
    __syncthreads();

    if (tid == 0) {
        const unsigned Kr = KTOP - cg;   // ranks still needed inside bucket q
        unsigned cum = 0, l = 0;
        for (int c = NCHUNK - 1; c >= 0; --c) {
            unsigned cs = csum[c];
            if (cum + cs >= Kr) {
                for (int b = c * 64 + 63; b >= c * 64; --b) {
                    unsigned h = hist[b >> 1];
                    unsigned cnt = (b & 1) ? (h >> 16) : (h & 0xFFFFu);
                    if (cum + cnt >= Kr) { l = (unsigned)b; break; }
                    cum += cnt;
                }
                break;
            }
            cum += cs;
        }
        thrArr[blockIdx.x] = (q << 16) | l;   // key of the K-th largest value
    }
}

// ---------------------------------------------------------------------
// Pass 3: fused WTA-mask + 5-step LIF recurrence.
// Stages x tiles via CDNA5 async global->LDS loads (asynccnt), reads
// back with ds_load_b128 (dscnt), all 5 time slices in flight at once.
// ---------------------------------------------------------------------
__global__ __launch_bounds__(256) void wta_lif(const float* __restrict__ x,
                                               const unsigned* __restrict__ thrArr,
                                               float* __restrict__ out) {
    __shared__ v4f stage[TS * 256];   // 20 KB staging buffer

    const int tid    = threadIdx.x;
    const int p4     = blockIdx.x * 256 + tid;   // float4 index within a sample
    const int binner = blockIdx.y;               // batch slot within a step

    // LDS byte base of `stage`: low 32 bits of the flat pointer (LDS aperture).
    const uint32_t sbase = (uint32_t)(uintptr_t)(void*)stage;

    // Issue all 5 async slice loads, then wait once on ASYNCcnt.
    #pragma unroll
    for (int t = 0; t < TS; ++t) {
        const size_t b = (size_t)(t * BS + binner);
        const v4f* gp = (const v4f*)x + b * (size_t)N4 + p4;
        const uint32_t loff = sbase + (uint32_t)((t * 256 + tid) * 16);
        asm volatile("global_load_async_to_lds_b128 %0, %1, off"
                     :: "v"(loff), "v"((unsigned long long)(uintptr_t)gp)
                     : "memory");
    }
    asm volatile("s_wait_asynccnt 0x0" ::: "memory");

    v4f u = {0.f, 0.f, 0.f, 0.f};
    #pragma unroll
    for (int t = 0; t < TS; ++t) {
        const uint32_t loff = sbase + (uint32_t)((t * 256 + tid) * 16);
        v4f v;
        asm volatile("ds_load_b128 %0, %1\n\ts_wait_dscnt 0x0"
                     : "=v"(v) : "v"(loff) : "memory");

        const int b = t * BS + binner;
        const unsigned thr = thrArr[b];          // uniform per block -> s_load

        // u = tau*u*(1 - spike(u_old)) + x_t ; s = spike(u_new) ; o = s * mask
        u.x = (u.x > 1.0f ? 0.0f : 0.5f * u.x) + v.x;
        u.y = (u.y > 1.0f ? 0.0f : 0.5f * u.y) + v.y;
        u.z = (u.z > 1.0f ? 0.0f : 0.5f * u.z) + v.z;
        u.w = (u.w > 1.0f ? 0.0f : 0.5f * u.w) + v.w;

        v4f o;
        o.x = ((u.x > 1.0f) && (fkey(v.x) >= thr)) ? 1.0f : 0.0f;
        o.y = ((u.y > 1.0f) && (fkey(v.y) >= thr)) ? 1.0f : 0.0f;
        o.z = ((u.z > 1.0f) && (fkey(v.z) >= thr)) ? 1.0f : 0.0f;
        o.w = ((u.w > 1.0f) && (fkey(v.w) >= thr)) ? 1.0f : 0.0f;

        ((v4f*)out)[(size_t)b * (size_t)N4 + p4] = o;
    }
}

// ---------------------------------------------------------------------
extern "C" void kernel_launch(void* const* d_in, const int* in_sizes, int n_in,
                              void* d_out, int out_size, void* d_ws, size_t ws_size,
                              hipStream_t stream) {
    const float* x = (const float*)d_in[0];
    float* out     = (float*)d_out;

    // workspace: [0,160) bucket ids, [160,320) count-above, [320,480) threshold keys
    int*      qArr   = (int*)d_ws;
    unsigned* cgArr  = (unsigned*)d_ws + NB;
    unsigned* thrArr = (unsigned*)d_ws + 2 * NB;

    wta_hist_hi<<<NB, 1024, 0, stream>>>(x, qArr, cgArr);
    wta_hist_lo<<<NB, 1024, 0, stream>>>(x, qArr, cgArr, thrArr);

    dim3 grid(CP / 1024, BS);   // 288 x 32 blocks, 256 threads (8 wave32s)
    wta_lif<<<grid, 256, 0, stream>>>(x, thrArr, out);
}